// global_softmax_pooling_44126493999222
// MI455X (gfx1250) — compile-verified
//
#include <hip/hip_runtime.h>

typedef __attribute__((ext_vector_type(2))) float v2f;
typedef __attribute__((ext_vector_type(8))) float v8f;

#define NEG_INF (-__builtin_inff())

// Kernel 1: segment start offsets via binary search (batch is sorted).
__global__ void seg_starts_kernel(const int* __restrict__ batch,
                                  int* __restrict__ starts, int N, int S) {
    int s = blockIdx.x * blockDim.x + threadIdx.x;
    if (s > S) return;
    if (s == S) { starts[S] = N; return; }
    int lo = 0, hi = N;
    while (lo < hi) {
        int mid = (lo + hi) >> 1;
        if (batch[mid] < s) lo = mid + 1; else hi = mid;
    }
    starts[s] = lo;
}

// Kernel 2: one block (16 wave32s) per segment.
// Single HBM pass: online softmax per channel, then WMMA row-reduction.
__global__ __launch_bounds__(512) void
segment_softmax_pool_kernel(const float* __restrict__ x,
                            const int* __restrict__ starts,
                            float* __restrict__ out) {
    const int s    = blockIdx.x;
    const int tid  = threadIdx.x;
    const int c4   = tid & 31;   // float4 column (channels 4*c4 .. 4*c4+3)
    const int r    = tid >> 5;   // row group 0..15 (== wave id)
    const int lane = tid & 31;

    const int beg = starts[s];
    const int end = starts[s + 1];
    const int cnt = end - beg;

    if (cnt == 0) {                       // block-uniform branch
        if (tid < 128) out[(size_t)s * 128 + tid] = 0.0f;
        return;
    }

    __shared__ float4 sA[16 * 32];        // running max panel, then sum(e) panel
    __shared__ float4 sB[16 * 32];        // sum(x*e) panel
    __shared__ float  smax[128];
    __shared__ float  red[256];

    // ---- single streaming pass: online softmax accumulation ----
    const float4* xv = reinterpret_cast<const float4*>(x);
    float4 m  = {NEG_INF, NEG_INF, NEG_INF, NEG_INF};
    float4 se = {0.f, 0.f, 0.f, 0.f};
    float4 sx = {0.f, 0.f, 0.f, 0.f};
    for (int row = beg + r; row < end; row += 16) {
        float4 v = xv[(size_t)row * 32 + c4];
#define ONLINE(c)                                             \
        { float mn = fmaxf(m.c, v.c);                         \
          float sc = __expf(m.c - mn);                        \
          float e  = __expf(v.c - mn);                        \
          se.c = se.c * sc + e;                               \
          sx.c = sx.c * sc + v.c * e;                         \
          m.c  = mn; }
        ONLINE(x) ONLINE(y) ONLINE(z) ONLINE(w)
#undef ONLINE
    }

    // ---- combine the 16 row-group maxima per channel ----
    sA[r * 32 + c4] = m;
    __syncthreads();
    if (tid < 128) {
        const float* pm = reinterpret_cast<const float*>(sA);
        float mm = pm[tid];
#pragma unroll
        for (int k = 1; k < 16; ++k) mm = fmaxf(mm, pm[k * 128 + tid]);
        smax[tid] = mm;
    }
    __syncthreads();

    // rescale partials to the common max, stage 16x128 panels for WMMA
    float4 mt = reinterpret_cast<const float4*>(smax)[c4];
    float4 pe, px;
#define RESCALE(c)                                            \
    { float sc = __expf(m.c - mt.c);                          \
      pe.c = se.c * sc;  px.c = sx.c * sc; }
    RESCALE(x) RESCALE(y) RESCALE(z) RESCALE(w)
#undef RESCALE
    sA[r * 32 + c4] = pe;
    sB[r * 32 + c4] = px;
    __syncthreads();

    // ---- reduce 16 rows per channel on the matrix pipe ----
    // A = ones (16x4 f32, 2 VGPRs), B = 4x16 slice of the panel
    // (VGPR0: K=0 lanes0-15 / K=2 lanes16-31; VGPR1: K=1 / K=3).
    // Four chained V_WMMA_F32_16X16X4_F32 sum all 16 rows; every D row
    // holds the column sums, D row M=0 lives in acc[0] lanes 0-15.
    const int q    = r >> 3;      // 0: sum(e) panel, 1: sum(x*e) panel
    const int tile = r & 7;       // 16-channel tile
    const float* P = (q == 0) ? reinterpret_cast<const float*>(sA)
                              : reinterpret_cast<const float*>(sB);
    const int col = tile * 16 + (lane & 15);
    const int hi  = lane >> 4;

    v2f a; a[0] = 1.0f; a[1] = 1.0f;
    v8f acc = {};
#pragma unroll
    for (int k = 0; k < 4; ++k) {
        v2f b;
        b[0] = P[(4 * k + 2 * hi + 0) * 128 + col];
        b[1] = P[(4 * k + 2 * hi + 1) * 128 + col];
        acc = __builtin_amdgcn_wmma_f32_16x16x4_f32(
            /*neg_a=*/false, a, /*neg_b=*/false, b,
            /*c_mod=*/(short)0, acc, /*reuse_a=*/false, /*reuse_b=*/false);
    }
    if (lane < 16) red[q * 128 + col] = acc[0];
    __syncthreads();

    // ---- finalize: weighted mean ----
    if (tid < 128) {
        float denom = red[tid] + 1e-16f;     // sum(e) + EPS
        float wsum  = red[128 + tid];        // sum(x*e)
        out[(size_t)s * 128 + tid] = wsum / denom / (float)cnt;
    }
}

extern "C" void kernel_launch(void* const* d_in, const int* in_sizes, int n_in,
                              void* d_out, int out_size, void* d_ws, size_t ws_size,
                              hipStream_t stream) {
    const float* x      = (const float*)d_in[0];
    const int*   batch  = (const int*)d_in[1];
    float*       out    = (float*)d_out;
    int*         starts = (int*)d_ws;        // (S+1) ints of scratch

    const int N = in_sizes[0] / 128;         // 500000
    const int S = out_size   / 128;          // 2048

    seg_starts_kernel<<<(S + 1 + 255) / 256, 256, 0, stream>>>(batch, starts, N, S);
    segment_softmax_pool_kernel<<<S, 512, 0, stream>>>(x, starts, out);
}